// TopKSoftmaxGate_89008902242849
// MI455X (gfx1250) — compile-verified
//
#include <hip/hip_runtime.h>

typedef __attribute__((ext_vector_type(2))) float v2f;
typedef __attribute__((ext_vector_type(8))) float v8f;

#define B_SZ 8192
#define D_SZ 512
#define E_SZ 16
#define DX_SZ 512
#define LDS_STRIDE 516   // 512 + 4 pad: lane l reads offset 4*l mod 64 banks -> conflict-free
#define SPARSITY_EPS 1e-5f

// ---------------------------------------------------------------------------
// Gate kernel: one wave (32 threads) per 16-row batch tile.
// logits = x @ W^T via V_WMMA_F32_16X16X4_F32, then per-row top-2 softmax,
// permutation mix, renormalize -> sp[b, 16] in workspace.
// ---------------------------------------------------------------------------
__global__ __launch_bounds__(32) void gate_kernel(
    const float* __restrict__ x,     // [B, 512]
    const float* __restrict__ W,     // [16, 512]
    const float* __restrict__ bias,  // [16]
    const float* __restrict__ pw,    // [4, 16, 16]
    float* __restrict__ spbuf)       // [B, 16]
{
  __shared__ float ldsx[16 * LDS_STRIDE];
  __shared__ float ldsw[16 * LDS_STRIDE];
  __shared__ float ldslog[16 * 17];
  __shared__ float ldsperm[16 * 17];

  const int tid = threadIdx.x;
  const int b0  = blockIdx.x * 16;

  // Stage x tile and W into LDS with coalesced float4 global loads.
  for (int i4 = tid; i4 < 16 * 128; i4 += 32) {
    const int r  = i4 >> 7;       // row 0..15
    const int c4 = i4 & 127;      // float4 column
    const float4 vx = ((const float4*)(x + (size_t)(b0 + r) * DX_SZ))[c4];
    const float4 vw = ((const float4*)(W + (size_t)r * DX_SZ))[c4];
    float* px = &ldsx[r * LDS_STRIDE + c4 * 4];
    float* pq = &ldsw[r * LDS_STRIDE + c4 * 4];
    px[0] = vx.x; px[1] = vx.y; px[2] = vx.z; px[3] = vx.w;
    pq[0] = vw.x; pq[1] = vw.y; pq[2] = vw.z; pq[3] = vw.w;
  }
  // Mean of permutation_weights over P=4 -> ldsperm[i][j]
  for (int i = tid; i < 256; i += 32) {
    float v = 0.25f * (pw[i] + pw[256 + i] + pw[512 + i] + pw[768 + i]);
    ldsperm[(i >> 4) * 17 + (i & 15)] = v;
  }
  __syncthreads();

  // WMMA accumulation: A = x tile (16xK), B = W^T (Kx16), D = logits 16x16.
  // f32 16x16x4 layout: lanes 0-15 hold K = k0,k0+1 ; lanes 16-31 hold K = k0+2,k0+3.
  const int half = tid >> 4;
  const int m    = tid & 15;
  const float* ax = &ldsx[m * LDS_STRIDE + 2 * half];
  const float* bw = &ldsw[m * LDS_STRIDE + 2 * half];
  v8f c = {};
  for (int k0 = 0; k0 < DX_SZ; k0 += 4) {
    v2f a = { ax[k0], ax[k0 + 1] };
    v2f b = { bw[k0], bw[k0 + 1] };
    c = __builtin_amdgcn_wmma_f32_16x16x4_f32(false, a, false, b,
                                              (short)0, c, false, false);
  }
  // C/D layout: VGPR r, lanes 0-15 -> M=r, lanes 16-31 -> M=r+8 ; N = lane%16.
#pragma unroll
  for (int r = 0; r < 8; ++r)
    ldslog[(r + 8 * half) * 17 + m] = c[r];
  __syncthreads();

  // Per-row gating (16 lanes, one batch row each).
  if (tid < 16) {
    float v[16];
#pragma unroll
    for (int e = 0; e < 16; ++e) v[e] = ldslog[tid * 17 + e] + bias[e];

    // top-2, ties resolved to lowest index (matches jax.lax.top_k)
    int i1 = 0; float w1 = v[0];
#pragma unroll
    for (int e = 1; e < 16; ++e) if (v[e] > w1) { w1 = v[e]; i1 = e; }
    int i2 = (i1 == 0) ? 1 : 0; float w2 = v[i2];
#pragma unroll
    for (int e = 0; e < 16; ++e)
      if (e != i1 && v[e] > w2) { w2 = v[e]; i2 = e; }

    // replicate scatter/zero->-inf quirk, then softmax over the two entries
    float m1 = (w1 != 0.0f) ? w1 : -__builtin_inff();
    float m2 = (w2 != 0.0f) ? w2 : -__builtin_inff();
    float mx = fmaxf(m1, m2);
    float e1 = __expf(m1 - mx);
    float e2 = __expf(m2 - mx);
    float inv = 1.0f / (e1 + e2);
    float s1 = e1 * inv, s2 = e2 * inv;

    // sp[j] = s1*perm[i1][j] + s2*perm[i2][j], then renormalize
    float sp[16]; float tot = 0.f;
#pragma unroll
    for (int j = 0; j < 16; ++j) {
      float pv = s1 * ldsperm[i1 * 17 + j] + s2 * ldsperm[i2 * 17 + j];
      sp[j] = pv; tot += pv;
    }
    float rinv = 1.0f / tot;
    float* outp = spbuf + (size_t)(b0 + tid) * 16;
#pragma unroll
    for (int j = 0; j < 16; j += 4) {
      float4 o = { sp[j] * rinv, sp[j+1] * rinv, sp[j+2] * rinv, sp[j+3] * rinv };
      ((float4*)outp)[j >> 2] = o;
    }
  }
}

// ---------------------------------------------------------------------------
// Combine kernel: y[b,d] = sum_e h[b,d,e] * sp[b,e]. Pure HBM streaming:
// one thread per (b,d) -> 4x global_load_b128 of contiguous e-values.
// ---------------------------------------------------------------------------
__global__ __launch_bounds__(256) void combine_kernel(
    const float* __restrict__ h,      // [B, D, 16]
    const float* __restrict__ spbuf,  // [B, 16]
    float* __restrict__ y)            // [B, D]
{
  const size_t t = (size_t)blockIdx.x * 256 + threadIdx.x;  // 0..B*D-1
  const size_t b = t >> 9;                                   // /D_SZ
  const float4* hp = (const float4*)(h + t * 16);
  const float4* sv = (const float4*)(spbuf + b * 16);
  float acc = 0.f;
#pragma unroll
  for (int q = 0; q < 4; ++q) {
    float4 hv = hp[q];
    float4 s  = sv[q];
    acc += hv.x * s.x + hv.y * s.y + hv.z * s.z + hv.w * s.w;
  }
  y[t] = acc;
}

// ---------------------------------------------------------------------------
// Averages kernel: deterministic single-block reduction over B.
// out[0:16] = mean(sp), out[16:32] = mean(sp >= eps ? 1 : 0).
// ---------------------------------------------------------------------------
__global__ __launch_bounds__(256) void avg_kernel(
    const float* __restrict__ spbuf, float* __restrict__ out)
{
  __shared__ float ssoft[256];
  __shared__ float shard[256];
  const int t = threadIdx.x;
  const int e = t & 15, g = t >> 4;   // 16 expert lanes x 16 groups
  float as = 0.f, ah = 0.f;
  for (int b = g; b < B_SZ; b += 16) {
    float v = spbuf[(size_t)b * 16 + e];
    as += v;
    ah += (v < SPARSITY_EPS) ? 0.f : 1.f;
  }
  ssoft[t] = as; shard[t] = ah;
  __syncthreads();
  if (t < 16) {
    float ts = 0.f, th = 0.f;
    for (int g2 = 0; g2 < 16; ++g2) { ts += ssoft[g2 * 16 + t]; th += shard[g2 * 16 + t]; }
    out[t]      = ts * (1.0f / B_SZ);
    out[16 + t] = th * (1.0f / B_SZ);
  }
}

extern "C" void kernel_launch(void* const* d_in, const int* in_sizes, int n_in,
                              void* d_out, int out_size, void* d_ws, size_t ws_size,
                              hipStream_t stream) {
  const float* h    = (const float*)d_in[0];  // [8192, 512, 16]
  const float* x    = (const float*)d_in[1];  // [8192, 512]
  const float* W    = (const float*)d_in[2];  // [16, 512]
  const float* bias = (const float*)d_in[3];  // [16]
  const float* pw   = (const float*)d_in[4];  // [4, 16, 16]
  // d_in[5] = k (== 2, hardcoded)

  float* out   = (float*)d_out;               // y | soft_avg | hard_avg
  float* spbuf = (float*)d_ws;                // [8192, 16] scratch

  gate_kernel<<<B_SZ / 16, 32, 0, stream>>>(x, W, bias, pw, spbuf);
  combine_kernel<<<(B_SZ * D_SZ) / 256, 256, 0, stream>>>(h, spbuf, out);
  avg_kernel<<<1, 256, 0, stream>>>(spbuf, out + (size_t)B_SZ * D_SZ);
}